// InfoNCEv2_59785944760612
// MI455X (gfx1250) — compile-verified
//
#include <hip/hip_runtime.h>
#include <math.h>

// ---------------------------------------------------------------------------
// InfoNCE-v2 loss for MI455X (gfx1250, wave32).
// GEMMs via V_WMMA_F32_16X16X4_F32 (exact f32 path) with a pre-packed,
// transposed+padded B so the inner loop is branch-free (no EXEC games):
//     global_load_b64 (A frag) + global_load_b64 (B frag) + v_wmma.
// Pairwise relu-inside-dot term is inherently VALU; done with LDS staging.
// ---------------------------------------------------------------------------

typedef __attribute__((ext_vector_type(2))) float v2f;
typedef __attribute__((ext_vector_type(8))) float v8f;

#define HID      300
#define HID_PAD  304
#define NROWS    2048
#define XD       512
#define CHUNK    256

__device__ __forceinline__ float softplusf(float x) {
  // numerically stable softplus
  return fmaxf(x, 0.0f) + log1pf(expf(-fabsf(x)));
}

// ---------------------------------------------------------------------------
// Pack B (K=512 x HID row-major) into Bt (HID_PAD x 512), transposed and
// zero-padded: Bt[c][k] = (c < HID) ? B[k][c] : 0.
// ---------------------------------------------------------------------------
__global__ __launch_bounds__(256)
void pack_bt_kernel(const float* __restrict__ B, float* __restrict__ Bt)
{
  const int idx = blockIdx.x * 256 + threadIdx.x;  // 0 .. HID_PAD*XD-1
  const int c = idx >> 9;    // / 512
  const int k = idx & 511;
  Bt[idx] = (c < HID) ? B[k * HID + c] : 0.0f;
}

// ---------------------------------------------------------------------------
// C(2048 x HID, ld HID_PAD) = A1(2048 x K1) @ Bt1^T  [+ A2 @ Bt2^T]
// One wave (32 threads) per 16x16 output tile; EXEC stays all-ones.
// A fragment (32-bit A 16x4): lane-half h, VGPR j -> (M = lane&15, K = k+2h+j)
// B fragment mirrors transposed:            VGPR j -> (K = k+2h+j, N = lane&15)
//   With Bt column-major, both fragments are contiguous float2 loads.
// D layout: VGPR r, lane-half h -> (M = r + 8h, N = lane&15).
// ---------------------------------------------------------------------------
__global__ __launch_bounds__(32)
void wmma_gemm_f32(const float* __restrict__ A1, const float* __restrict__ Bt1, int K1,
                   const float* __restrict__ A2, const float* __restrict__ Bt2, int K2,
                   float* __restrict__ C)
{
  const int mt   = blockIdx.x;
  const int nt   = blockIdx.y;
  const int lane = threadIdx.x;      // 0..31, full wave always active
  const int half = lane >> 4;        // 0 or 1
  const int lidx = lane & 15;

  const float* arow1 = A1  + (mt * 16 + lidx) * XD + 2 * half;
  const float* brow1 = Bt1 + (nt * 16 + lidx) * XD + 2 * half;
  __builtin_prefetch(arow1, 0, 0);
  __builtin_prefetch(brow1, 0, 0);

  v8f acc = {};
  for (int k = 0; k < K1; k += 4) {
    v2f a = *(const v2f*)(arow1 + k);
    v2f b = *(const v2f*)(brow1 + k);
    acc = __builtin_amdgcn_wmma_f32_16x16x4_f32(false, a, false, b,
                                                (short)0, acc, false, false);
  }
  if (K2 > 0) {
    const float* arow2 = A2  + (mt * 16 + lidx) * XD + 2 * half;
    const float* brow2 = Bt2 + (nt * 16 + lidx) * XD + 2 * half;
    for (int k = 0; k < K2; k += 4) {
      v2f a = *(const v2f*)(arow2 + k);
      v2f b = *(const v2f*)(brow2 + k);
      acc = __builtin_amdgcn_wmma_f32_16x16x4_f32(false, a, false, b,
                                                  (short)0, acc, false, false);
    }
  }

  const float* accf = (const float*)&acc;
#pragma unroll
  for (int r = 0; r < 8; ++r) {
    C[(mt * 16 + r + 8 * half) * HID_PAD + nt * 16 + lidx] = accf[r];
  }
}

// ---------------------------------------------------------------------------
// t0 term: srow[row] = softplus(relu(POS[row,:] + b1) . w2 + b2)
// 8 waves per block, one row per wave, wave32 shuffle reduction.
// ---------------------------------------------------------------------------
__global__ __launch_bounds__(256)
void t0_rows_kernel(const float* __restrict__ POS, const float* __restrict__ b1,
                    const float* __restrict__ w2, const float* __restrict__ b2,
                    float* __restrict__ srow)
{
  const int wave = threadIdx.x >> 5;
  const int lane = threadIdx.x & 31;
  const int row  = blockIdx.x * 8 + wave;
  const float* p = POS + row * HID_PAD;

  float partial = 0.0f;
  for (int k = lane; k < HID; k += 32) {
    partial = fmaf(fmaxf(p[k] + b1[k], 0.0f), w2[k], partial);
  }
#pragma unroll
  for (int off = 16; off > 0; off >>= 1)
    partial += __shfl_down(partial, off, 32);
  if (lane == 0) srow[row] = softplusf(partial + b2[0]);
}

// ---------------------------------------------------------------------------
// Pairwise term. One block per global g-row (2048 total). Thread t owns
// l = chunk_base + t. GP row (+b1) and w2 staged in LDS; block-wide
// logsumexp over the 256 softplus scores.
// ---------------------------------------------------------------------------
__global__ __launch_bounds__(256)
void pair_lse_kernel(const float* __restrict__ LP, const float* __restrict__ GP,
                     const float* __restrict__ b1, const float* __restrict__ w2,
                     const float* __restrict__ b2, float* __restrict__ lse)
{
  __shared__ float sg[HID_PAD];
  __shared__ float sw[HID_PAD];
  __shared__ float red[256];

  const int g  = blockIdx.x;          // 0..2047
  const int t  = threadIdx.x;         // 0..255 == l within chunk
  const int l0 = (g >> 8) << 8;       // chunk base row

  for (int k = t; k < HID; k += 256) {
    sg[k] = GP[g * HID_PAD + k] + b1[k];
    sw[k] = w2[k];
  }
  __syncthreads();

  const float* lp = LP + (l0 + t) * HID_PAD;
  float acc = 0.0f;
  for (int k = 0; k < HID; ++k) {
    float h = fmaxf(sg[k] + lp[k], 0.0f);
    acc = fmaf(h, sw[k], acc);
  }
  const float s = softplusf(acc + b2[0]);

  // logsumexp over the 256 scores
  red[t] = s;
  __syncthreads();
  for (int str = 128; str > 0; str >>= 1) {
    if (t < str) red[t] = fmaxf(red[t], red[t + str]);
    __syncthreads();
  }
  const float m = red[0];
  __syncthreads();
  red[t] = expf(s - m);
  __syncthreads();
  for (int str = 128; str > 0; str >>= 1) {
    if (t < str) red[t] += red[t + str];
    __syncthreads();
  }
  if (t == 0) lse[g] = m + logf(red[0]);
}

// ---------------------------------------------------------------------------
// Deterministic final reduction: out = (sum(lse) - sum(srow)) / 2048
//   ( == t1 - t0 == -(t0 - t1) )
// ---------------------------------------------------------------------------
__global__ __launch_bounds__(256)
void finalize_kernel(const float* __restrict__ srow, const float* __restrict__ lse,
                     float* __restrict__ out)
{
  __shared__ float red[256];
  const int t = threadIdx.x;
  float a = 0.0f;
  for (int i = t; i < NROWS; i += 256) a += lse[i] - srow[i];
  red[t] = a;
  __syncthreads();
  for (int str = 128; str > 0; str >>= 1) {
    if (t < str) red[t] += red[t + str];
    __syncthreads();
  }
  if (t == 0) out[0] = red[0] / (float)NROWS;
}

extern "C" void kernel_launch(void* const* d_in, const int* in_sizes, int n_in,
                              void* d_out, int out_size, void* d_ws, size_t ws_size,
                              hipStream_t stream) {
  (void)in_sizes; (void)n_in; (void)out_size; (void)ws_size;

  const float* pos_local  = (const float*)d_in[0];   // 2048 x 512
  const float* pos_global = (const float*)d_in[1];   // 2048 x 512
  const float* neg_local  = (const float*)d_in[2];   // 2048 x 512
  const float* neg_global = (const float*)d_in[3];   // 2048 x 512
  const float* W1         = (const float*)d_in[4];   // 1024 x 300 row-major
  const float* b1         = (const float*)d_in[5];   // 300
  const float* W2         = (const float*)d_in[6];   // 300 x 1
  const float* b2         = (const float*)d_in[7];   // 1
  // d_in[8] = batch_size (==256, baked into CHUNK)

  const float* W1l = W1;                 // rows 0..511
  const float* W1g = W1 + XD * HID;      // rows 512..1023

  float* POS  = (float*)d_ws;            // 2048 x 304
  float* LP   = POS  + NROWS * HID_PAD;  // 2048 x 304
  float* GP   = LP   + NROWS * HID_PAD;  // 2048 x 304
  float* srow = POS  + 3 * NROWS * HID_PAD;  // 2048
  float* lse  = srow + NROWS;            // 2048
  float* BtL  = lse  + NROWS;            // 304 x 512 (W1l transposed, padded)
  float* BtG  = BtL  + HID_PAD * XD;     // 304 x 512 (W1g transposed, padded)

  // Pack / transpose / zero-pad the weight matrices.
  pack_bt_kernel<<<(HID_PAD * XD) / 256, 256, 0, stream>>>(W1l, BtL);
  pack_bt_kernel<<<(HID_PAD * XD) / 256, 256, 0, stream>>>(W1g, BtG);

  dim3 ggrid(NROWS / 16, HID_PAD / 16);  // 128 x 19 tiles

  // POS = pos_local @ W1l + pos_global @ W1g
  wmma_gemm_f32<<<ggrid, 32, 0, stream>>>(pos_local, BtL, XD,
                                          pos_global, BtG, XD, POS);
  // LP = neg_local @ W1l
  wmma_gemm_f32<<<ggrid, 32, 0, stream>>>(neg_local, BtL, XD,
                                          nullptr, nullptr, 0, LP);
  // GP = neg_global @ W1g
  wmma_gemm_f32<<<ggrid, 32, 0, stream>>>(neg_global, BtG, XD,
                                          nullptr, nullptr, 0, GP);

  t0_rows_kernel<<<NROWS / 8, 256, 0, stream>>>(POS, b1, W2, b2, srow);
  pair_lse_kernel<<<NROWS, 256, 0, stream>>>(LP, GP, b1, W2, b2, lse);
  finalize_kernel<<<1, 256, 0, stream>>>(srow, lse, (float*)d_out);
}